// HashEncoder_66228395704407
// MI455X (gfx1250) — compile-verified
//
#include <hip/hip_runtime.h>
#include <stdint.h>

typedef float v2f __attribute__((ext_vector_type(2)));

#define PRIME1 2654435761u
#define PRIME2 805459861u
#define HMASK  0x7FFFFu   // hashed levels all have size 2^19 exactly

// Block = 256 threads = 16 points x 16 levels.
__global__ __launch_bounds__(256) void hashgrid_fwd(
    const float* __restrict__ in,   // [N,3]
    const float* __restrict__ emb,  // [TOTAL,2]
    float* __restrict__ out,        // [N,32]
    int npts)
{
    __shared__ float s_xyz[48];

    const int tid = threadIdx.x;
    const long long pt0 = (long long)blockIdx.x * 16;

    // CDNA5 async global->LDS copy of this block's coords (48 floats).
    if (tid < 48 && (pt0 * 3 + tid) < (long long)npts * 3) {
        const float* g = in + pt0 * 3 + tid;
        __builtin_amdgcn_global_load_async_to_lds_b32(
            (__attribute__((address_space(1))) int*)g,
            (__attribute__((address_space(3))) int*)(s_xyz + tid),
            /*offset=*/0, /*cpol=*/0);
    }
    __builtin_amdgcn_s_wait_asynccnt(0);
    __syncthreads();

    const int lp    = tid >> 4;   // point within block
    const int level = tid & 15;   // hash level
    const long long p = pt0 + lp;
    if (p >= npts) return;

    // normalization: x01 = (x + 1) / 2
    const float x = (s_xyz[lp * 3 + 0] + 1.0f) * 0.5f;
    const float y = (s_xyz[lp * 3 + 1] + 1.0f) * 0.5f;
    const float z = (s_xyz[lp * 3 + 2] + 1.0f) * 0.5f;

    const float rf = (float)(16 << level);
    const float px = x * rf, py = y * rf, pz = z * rf;
    const float gx = floorf(px), gy = floorf(py), gz = floorf(pz);
    const float fx = px - gx, fy = py - gy, fz = pz - gz;
    const uint32_t ix = (uint32_t)gx, iy = (uint32_t)gy, iz = (uint32_t)gz;

    const uint32_t s1 = (16u << level) + 1u;   // r+1
    const uint32_t s2 = s1 * s1;               // (r+1)^2
    const bool dense = (level < 3);
    const uint32_t off = (level == 0) ? 0u
                       : (level == 1) ? 4913u
                       : (level == 2) ? 40850u
                       : 315475u + ((uint32_t)(level - 3) << 19);
    const v2f* __restrict__ tbl = (const v2f*)emb + off;

    // Branch-free index computation for all 8 corners.
    uint32_t idx[8];
#pragma unroll
    for (int c = 0; c < 8; ++c) {
        const uint32_t cx = ix + (c & 1);
        const uint32_t cy = iy + ((c >> 1) & 1);
        const uint32_t cz = iz + ((c >> 2) & 1);
        const uint32_t lin = cx + cy * s1 + cz * s2;
        const uint32_t h   = (cx ^ (cy * PRIME1) ^ (cz * PRIME2)) & HMASK;
        idx[c] = dense ? lin : h;
    }

    // 8 independent b64 gathers (L2-resident table, RT hint).
    v2f v[8];
#pragma unroll
    for (int c = 0; c < 8; ++c) v[c] = tbl[idx[c]];

    const float wx[2] = {1.0f - fx, fx};
    const float wy[2] = {1.0f - fy, fy};
    const float wz[2] = {1.0f - fz, fz};
    float a0 = 0.0f, a1 = 0.0f;
#pragma unroll
    for (int c = 0; c < 8; ++c) {
        const float w = wx[c & 1] * wy[(c >> 1) & 1] * wz[(c >> 2) & 1];
        a0 = fmaf(w, v[c].x, a0);
        a1 = fmaf(w, v[c].y, a1);
    }

    // Coalesced (256B/wave) non-temporal store: don't evict the table from L2.
    v2f r; r.x = a0; r.y = a1;
    __builtin_nontemporal_store(r, (v2f*)out + p * 16 + level);
}

extern "C" void kernel_launch(void* const* d_in, const int* in_sizes, int n_in,
                              void* d_out, int out_size, void* d_ws, size_t ws_size,
                              hipStream_t stream) {
    (void)n_in; (void)out_size; (void)d_ws; (void)ws_size;
    const float* in  = (const float*)d_in[0];   // [B,3] fp32
    const float* emb = (const float*)d_in[1];   // [TOTAL,2] fp32
    float* out = (float*)d_out;                 // [B,32] fp32

    const int npts = in_sizes[0] / 3;           // B = 2^21
    const int blocks = (npts + 15) / 16;        // 16 points per block
    hashgrid_fwd<<<blocks, 256, 0, stream>>>(in, emb, out, npts);
}